// Primitive_51316269252614
// MI455X (gfx1250) — compile-verified
//
#include <hip/hip_runtime.h>

// ---------------------------------------------------------------------------
// 2x2 gate on qubit axis 10 of a 24-qubit statevector, batch 4 (fp32).
// state = (L=1024, T=2, R=32768) floats contiguous in R.
// Pair stride = 32768 floats (131072 bytes), slice period = 65536 floats.
//
// Memory-bound streaming kernel (AI = 0.375 FLOP/B -> ~22us at 23.3 TB/s).
// WMMA deliberately NOT used: the only f32 shape (16x16x4) gives <=12.5% MAC
// utilization for a 2x2 gate and scatters D into lanes 0-3/16-19 across 8
// VGPRs, forcing cross-lane permutes costlier than the 8 FMAs replaced.
// Instead we use the CDNA5 async data mover: global_load_async_to_lds_b128
// (GVS form: SGPR base + 32-bit VGPR byte offset + 24-bit imm offset for the
// pair partner), double-buffered per wave, fenced with s_wait_asynccnt.
// Non-temporal policy on both streams (512MB touched once >> 192MB L2).
// ---------------------------------------------------------------------------

#define HALF_V4    8192u     // 32768 floats / 4  (pair distance, float4 units)
#define PERIOD_V4  16384u    // 65536 floats / 4  (one L-slice, float4 units)

typedef float v4f __attribute__((ext_vector_type(4)));

// Issue both halves of one tile: half0 at voff, half1 at voff + 131072 bytes
// (24-bit instruction immediate). LDS slots 4096 bytes apart.
__device__ __forceinline__ void async_ld_pair(unsigned lds_half0,
                                              unsigned voff_bytes,
                                              const void* sbase) {
  asm volatile(
      "global_load_async_to_lds_b128 %0, %2, %3 th:TH_LOAD_NT\n\t"
      "global_load_async_to_lds_b128 %1, %2, %3 offset:131072 th:TH_LOAD_NT"
      :
      : "v"(lds_half0), "v"(lds_half0 + 4096u), "v"(voff_bytes), "s"(sbase)
      : "memory");
}

__device__ __forceinline__ unsigned pair_v4_index(unsigned q) {
  // q = global float4-pair index; returns float4 index of half0.
  return (q >> 13) * PERIOD_V4 + (q & 8191u);
}

__global__ __launch_bounds__(256) void pauli_gate_async(
    const float* __restrict__ in, const float* __restrict__ P,
    v4f* __restrict__ out, unsigned T /* total threads */) {
  constexpr int NT = 4;               // float4-pairs per thread
  __shared__ v4f smem[2][2][256];     // [buf][half][thread] = 16 KB

  const float p00 = P[0], p01 = P[1], p10 = P[2], p11 = P[3];
  const unsigned tid = blockIdx.x * 256u + threadIdx.x;

  // Low 32 bits of a flat LDS pointer are the LDS byte offset on gfx1250.
  const unsigned lds0 = (unsigned)(size_t)(const void*)&smem[0][0][threadIdx.x];
  // slot(buf,half) = lds0 + buf*8192 + half*4096

  // Prologue: tile 0 -> buffer 0.
  async_ld_pair(lds0, pair_v4_index(tid) * 16u, (const void*)in);

  unsigned q = tid;
#pragma unroll
  for (int t = 0; t < NT; ++t) {
    const int buf = t & 1;

    if (t + 1 < NT) {
      // Issue tile t+1 into the other buffer, then wait for tile t
      // (outstanding 4 -> wait <=2 means tile t's two loads landed).
      async_ld_pair(lds0 + (unsigned)((t + 1) & 1) * 8192u,
                    pair_v4_index(tid + (unsigned)(t + 1) * T) * 16u,
                    (const void*)in);
      asm volatile("s_wait_asynccnt 2" ::: "memory");
    } else {
      asm volatile("s_wait_asynccnt 0" ::: "memory");
    }

    const v4f a = smem[buf][0][threadIdx.x];   // ds_load_b128
    const v4f b = smem[buf][1][threadIdx.x];

    const v4f y0 = p00 * a + p01 * b;          // packed mul + fma
    const v4f y1 = p10 * a + p11 * b;

    v4f* o0 = out + pair_v4_index(q);
    __builtin_nontemporal_store(y0, o0);               // store b128, NT
    __builtin_nontemporal_store(y1, o0 + HALF_V4);     // store b128, NT

    q += T;
  }
}

// Fallback (generic sizes / safety): plain streaming kernel, element pairs.
__global__ __launch_bounds__(256) void pauli_gate_direct(
    const float* __restrict__ in, const float* __restrict__ P,
    float* __restrict__ out, long long nPairs) {
  const float p00 = P[0], p01 = P[1], p10 = P[2], p11 = P[3];
  long long i = (long long)blockIdx.x * blockDim.x + threadIdx.x;
  const long long gs = (long long)gridDim.x * blockDim.x;
  for (; i < nPairs; i += gs) {
    const long long slice = i >> 15, r = i & 32767;
    const long long i0 = slice * 65536 + r;
    const float s0 = in[i0];
    const float s1 = in[i0 + 32768];
    out[i0]         = fmaf(p00, s0, p01 * s1);
    out[i0 + 32768] = fmaf(p10, s0, p11 * s1);
  }
}

extern "C" void kernel_launch(void* const* d_in, const int* in_sizes, int n_in,
                              void* d_out, int out_size, void* d_ws,
                              size_t ws_size, hipStream_t stream) {
  (void)in_sizes; (void)n_in; (void)d_ws; (void)ws_size;
  const float* state = (const float*)d_in[0];
  const float* pauli = (const float*)d_in[1];

  const long long total = (long long)out_size;   // floats (2^26 here)
  const long long pairsV4 = total / 8;           // float4 pairs
  constexpr int NT = 4;
  const long long T = pairsV4 / NT;              // total threads, async path

  if (total % 65536 == 0 && pairsV4 % NT == 0 && T % 256 == 0 &&
      T <= 0x7FFFFFFFll) {
    // 2^26 floats -> T = 2,097,152 threads -> 8192 blocks of 256.
    dim3 grid((unsigned)(T / 256)), block(256);
    pauli_gate_async<<<grid, block, 0, stream>>>(
        state, pauli, (v4f*)d_out, (unsigned)T);
  } else {
    const long long nPairs = total / 2;
    long long blocks = (nPairs + 255) / 256;
    if (blocks > (1ll << 16)) blocks = 1ll << 16;
    pauli_gate_direct<<<dim3((unsigned)blocks), dim3(256), 0, stream>>>(
        state, pauli, (float*)d_out, nPairs);
  }
}